// Classification2_10651518894900
// MI455X (gfx1250) — compile-verified
//
#include <hip/hip_runtime.h>

// Problem constants (reference: SEQ_LEN=128, FRAME=3x224x224)
#define S       128
#define D       150528           // 3*224*224
#define CHUNK   384              // K per block (split-K); 392*384 = 150528
#define KSPLIT  (D / CHUNK)      // 392 blocks
#define SLAB    32               // K staged into LDS per buffer
#define NSTEP   (SLAB / 4)       // 8 WMMA k-steps per slab
#define NSLAB   (CHUNK / SLAB)   // 12 slabs per block
#define LDSPITCH 36              // floats per B column in LDS (144B: 16B aligned,
                                 // conflict-free b64 fragment reads)

typedef __attribute__((ext_vector_type(2))) float v2f;
typedef __attribute__((ext_vector_type(8))) float v8f;

// Workspace layout (floats):
//   [0, 256)          : sq_a (clip2 norms) | sq_b (clip1 norms)
//   [256, 16640)      : gram (final, f32)
//   [16640, ...)      : per-block partials [KSPLIT][128*128] (if ws_size allows)

// Generic pointer -> LDS byte offset. AMDGPU lowers the AS3->flat cast as
// {shared_aperture_hi32, lds_offset_lo32}, so the low 32 bits are the DS offset.
__device__ __forceinline__ unsigned lds_byte_off(const void* p) {
    return (unsigned)(uintptr_t)p;
}

// Async-DMA one B column slab segment (8 x 16B) into LDS. The instruction adds
// nothing we need between chunks that differs on the two sides: within a column
// LDS and global both advance 16B per chunk.
__device__ __forceinline__ void async_stage_col(const float* gsrc, unsigned lds) {
    #pragma unroll
    for (int i = 0; i < 8; ++i) {
        asm volatile("global_load_async_to_lds_b128 %0, %1, off"
                     :: "v"(lds + 16u * (unsigned)i), "v"(gsrc + 4 * i)
                     : "memory");
    }
}

__device__ __forceinline__ void wait_async_all() {
    asm volatile("s_wait_asynccnt 0x0" ::: "memory");
}

// -------------------------------------------------------------------------
__global__ void zero_gram(float* __restrict__ gram) {
    gram[blockIdx.x * blockDim.x + threadIdx.x] = 0.0f;
}

// One block (256 threads) per row; blocks 0..127 -> clip2 (sq_a), 128..255 -> clip1 (sq_b).
__global__ void row_norms(const float* __restrict__ clip1,
                          const float* __restrict__ clip2,
                          float* __restrict__ sq) {
    const int row = blockIdx.x & (S - 1);
    const float* src = (blockIdx.x < S ? clip2 : clip1) + (size_t)row * D;
    const float4* v = (const float4*)src;
    float s = 0.0f;
    for (int i = threadIdx.x; i < D / 4; i += 256) {
        float4 x = v[i];
        s += x.x * x.x + x.y * x.y + x.z * x.z + x.w * x.w;
    }
    for (int off = 16; off; off >>= 1) s += __shfl_xor(s, off, 32);
    __shared__ float ls[8];
    if ((threadIdx.x & 31) == 0) ls[threadIdx.x >> 5] = s;
    __syncthreads();
    if (threadIdx.x == 0) {
        float t = 0.0f;
        #pragma unroll
        for (int w = 0; w < 8; ++w) t += ls[w];
        sq[blockIdx.x] = t;
    }
}

// -------------------------------------------------------------------------
// Split-K Gram: grid = (1, KSPLIT), block = 128 (4 waves).
// Block covers the full 128x128 output; wave w owns row strips 2w, 2w+1
// (16 v8f accumulators = 128 VGPRs). B chunk is async-DMA'd into
// double-buffered LDS (ASYNCcnt) and shared by all 4 waves -> global traffic
// is the 154MB minimum (A once, B once). A fragments stream from global.
//
// f32 WMMA fragment layout (16x4 A / 4x16 B, 2 VGPRs):
//   lanes 0-15  : K = k0,k0+1 for row/col (lane&15)
//   lanes 16-31 : K = k0+2,k0+3
template <int USE_ATOMIC>
__global__ __launch_bounds__(128)
void gram_wmma(const float* __restrict__ a,    // clip2 [S,D] -> rows x
               const float* __restrict__ b,    // clip1 [S,D] -> cols y
               float* __restrict__ dst) {      // gram (atomic) or partials
    __shared__ __align__(16) float bs[2][S * LDSPITCH];   // 2 x 18KB

    const int tid  = threadIdx.x;
    const int wave = tid >> 5;
    const int lane = tid & 31;
    const int half = lane >> 4;
    const int r    = lane & 15;
    const int kbase = blockIdx.y * CHUNK;

    // A row pointers for this wave's two strips (rows wave*32 + r, +16)
    const float* arow0 = a + (size_t)(wave * 32 + r) * D + 2 * half;
    const float* arow1 = arow0 + (size_t)16 * D;
    // staging: thread t DMA-streams B column (= clip1 row) t
    const float* brow = b + (size_t)tid * D;
    const unsigned ldscol0 = lds_byte_off(&bs[0][tid * LDSPITCH]);
    const unsigned ldscol1 = lds_byte_off(&bs[1][tid * LDSPITCH]);

    v8f acc[2][8] = {};

    // prologue: DMA slab 0 into buffer 0
    async_stage_col(brow + kbase, ldscol0);
    wait_async_all();
    __syncthreads();

    for (int s = 0; s < NSLAB; ++s) {
        const int kslab = kbase + s * SLAB;

        // kick DMA for the next slab; it lands under the 128 WMMAs below
        if (s + 1 < NSLAB)
            async_stage_col(brow + kslab + SLAB, ((s + 1) & 1) ? ldscol1 : ldscol0);

        const float* bbuf = &bs[s & 1][0];
        #pragma unroll
        for (int t = 0; t < NSTEP; ++t) {
            const int k    = kslab + 4 * t;
            const int koff = 4 * t + 2 * half;
            v2f a0 = *(const v2f*)(arow0 + k);
            v2f a1 = *(const v2f*)(arow1 + k);
            #pragma unroll
            for (int n = 0; n < 8; ++n) {
                v2f bf = *(const v2f*)&bbuf[(n * 16 + r) * LDSPITCH + koff];
                acc[0][n] = __builtin_amdgcn_wmma_f32_16x16x4_f32(
                    false, a0, false, bf, (short)0, acc[0][n], false, false);
                acc[1][n] = __builtin_amdgcn_wmma_f32_16x16x4_f32(
                    false, a1, false, bf, (short)0, acc[1][n], false, false);
            }
        }

        wait_async_all();        // own DMA done
        __syncthreads();         // everyone done reading buf s & sees staged s+1
    }

    // writeout: C/D layout -> VGPR v of lane (half,r) = element (M=v+8*half, N=r)
    #pragma unroll
    for (int sl = 0; sl < 2; ++sl) {
        const int m0 = (wave * 2 + sl) * 16;
        #pragma unroll
        for (int n = 0; n < 8; ++n) {
            #pragma unroll
            for (int v = 0; v < 8; ++v) {
                const int M = m0 + v + 8 * half;
                const int N = n * 16 + r;
                if (USE_ATOMIC) {
                    atomicAdd(&dst[M * S + N], acc[sl][n][v]);
                } else {
                    dst[(size_t)blockIdx.y * (S * S) + M * S + N] = acc[sl][n][v];
                }
            }
        }
    }
}

// Sum KSPLIT partials into gram. 64 blocks x 256 threads; coalesced reads.
__global__ void reduce_partials(const float* __restrict__ partials,
                                float* __restrict__ gram) {
    const int e = blockIdx.x * 256 + threadIdx.x;
    float s = 0.0f;
    for (int j = 0; j < KSPLIT; ++j) s += partials[(size_t)j * (S * S) + e];
    gram[e] = s;
}

// -------------------------------------------------------------------------
// One block per output (129 diagonals); thread x handles matrix[x, x+d].
__global__ void diag_avg(const float* __restrict__ gram,
                         const float* __restrict__ sq,   // [256]: sq_a | sq_b
                         float* __restrict__ out) {      // [S+1]
    const int t = blockIdx.x;           // result index 63 + t
    const int d = t - (S / 2);          // diagonal offset y - x in [-64, 64]
    const int x = threadIdx.x;
    const int y = x + d;
    float val = 0.0f;
    if (y >= 0 && y < S) {
        const float sa = sq[x];         // ||clip2_x||^2
        const float sb = sq[S + y];     // ||clip1_y||^2
        const float g  = gram[x * S + y];
        val = -((sa + sb - 2.0f * g) * (1.0e13f / (float)D));
    }
    for (int off = 16; off; off >>= 1) val += __shfl_xor(val, off, 32);
    __shared__ float ls[4];
    if ((threadIdx.x & 31) == 0) ls[threadIdx.x >> 5] = val;
    __syncthreads();
    if (threadIdx.x == 0) {
        const float sum = ls[0] + ls[1] + ls[2] + ls[3];
        const int count = S - (d < 0 ? -d : d);
        out[t] = sum / (float)count;
    }
}

// -------------------------------------------------------------------------
extern "C" void kernel_launch(void* const* d_in, const int* in_sizes, int n_in,
                              void* d_out, int out_size, void* d_ws, size_t ws_size,
                              hipStream_t stream) {
    const float* clip1 = (const float*)d_in[0];
    const float* clip2 = (const float*)d_in[1];
    float* out   = (float*)d_out;
    float* ws    = (float*)d_ws;
    float* sq    = ws;                // 256 floats
    float* gram  = ws + 256;          // 16384 floats
    float* parts = ws + 16640;        // KSPLIT * 16384 floats (optional)

    const size_t need_bytes =
        ((size_t)16640 + (size_t)KSPLIT * S * S) * sizeof(float);
    const bool use_partials = (ws_size >= need_bytes);

    row_norms<<<2 * S, 256, 0, stream>>>(clip1, clip2, sq);

    if (use_partials) {
        gram_wmma<0><<<dim3(1, KSPLIT), 128, 0, stream>>>(clip2, clip1, parts);
        reduce_partials<<<S * S / 256, 256, 0, stream>>>(parts, gram);
    } else {
        zero_gram<<<S * S / 256, 256, 0, stream>>>(gram);
        gram_wmma<1><<<dim3(1, KSPLIT), 128, 0, stream>>>(clip2, clip1, gram);
    }

    diag_avg<<<S + 1, 128, 0, stream>>>(gram, sq, out);
}